// GraphGAT_3839700762921
// MI455X (gfx1250) — compile-verified
//
#include <hip/hip_runtime.h>
#include <hip/hip_bf16.h>
#include <math.h>

typedef __attribute__((ext_vector_type(2))) float v2f;
typedef __attribute__((ext_vector_type(8))) float v8f;

#define B_  16
#define C_  2048
#define L_  256
#define EO  16
#define N_  (B_*C_)
#define NE_ 524288

// ---- monotonic float<->uint mapping for atomicMax on signed floats ----
__device__ __forceinline__ unsigned fmap(float f) {
  unsigned u = __float_as_uint(f);
  return (u & 0x80000000u) ? ~u : (u | 0x80000000u);
}
__device__ __forceinline__ float funmap(unsigned u) {
  return __uint_as_float((u & 0x80000000u) ? (u ^ 0x80000000u) : ~u);
}
#define FMAP_NEG_INF 0x007FFFFFu   // fmap(-inf)

// ---------------- Kernel 1: x = feature @ emb_W.T + emb_b, accumulate zsum
// one wave per 16-row tile, K=256 in steps of 4 via v_wmma_f32_16x16x4_f32
__global__ __launch_bounds__(256) void k_embed(const float* __restrict__ feature,
    const float* __restrict__ emb_W, const float* __restrict__ emb_b,
    float* __restrict__ x, float* __restrict__ zsum) {
  int wave = (blockIdx.x * blockDim.x + threadIdx.x) >> 5;   // global tile id
  int lane = threadIdx.x & 31;
  int i0 = wave * 16;
  int M  = lane & 15;
  int hi = lane >> 4;
  const float* arow = feature + (size_t)(i0 + M) * L_;
  const float* brow = emb_W  + (size_t)M * L_;               // col == lane&15
  v8f c = {};
  for (int kb = 0; kb < L_; kb += 4) {
    int k = kb + 2 * hi;
    v2f a, b;
    a.x = arow[k];  a.y = arow[k + 1];
    b.x = brow[k];  b.y = brow[k + 1];
    c = __builtin_amdgcn_wmma_f32_16x16x4_f32(false, a, false, b,
                                              (short)0, c, false, false);
  }
  float bias = emb_b[M];
  float zpart = 0.f;
  for (int v = 0; v < 8; ++v) {
    int row = i0 + v + 8 * hi;
    float val = c[v] + bias;
    x[(size_t)row * EO + M] = val;
    zpart += val;
  }
  atomicAdd(&zsum[(i0 >> 11) * EO + M], zpart);
}

// ---------------- Kernel 2: SE block (tiny)
__global__ __launch_bounds__(256) void k_se(const float* __restrict__ zsum,
    const float* __restrict__ W1, const float* __restrict__ b1,
    const float* __restrict__ W2, const float* __restrict__ b2,
    float* __restrict__ s_out) {
  __shared__ float z[B_ * EO];
  __shared__ float r1[B_ * 4];
  int t = threadIdx.x;
  if (t < B_ * EO) z[t] = zsum[t] * (1.0f / C_);
  __syncthreads();
  if (t < B_ * 4) {
    int b = t >> 2, r = t & 3;
    float acc = b1[r];
    for (int e = 0; e < EO; ++e) acc += z[b * EO + e] * W1[r * EO + e];
    r1[t] = acc > 0.f ? acc : 0.f;
  }
  __syncthreads();
  if (t < B_ * EO) {
    int b = t >> 4, e = t & 15;
    float acc = b2[e];
    for (int r = 0; r < 4; ++r) acc += r1[b * 4 + r] * W2[e * 4 + r];
    s_out[t] = 1.0f / (1.0f + __expf(-acc));
  }
}

// ---------------- Kernel 3: feat = (x*s) @ gat_W.T, el, er; init edge buffers
__global__ __launch_bounds__(256) void k_feat(const float* __restrict__ x,
    const float* __restrict__ s, const float* __restrict__ gat_W,
    const float* __restrict__ attn_l, const float* __restrict__ attn_r,
    float* __restrict__ feat, float* __restrict__ el, float* __restrict__ er,
    unsigned* __restrict__ mbuf, float* __restrict__ denom,
    float* __restrict__ accum) {
  int n = blockIdx.x * blockDim.x + threadIdx.x;
  if (n >= N_) return;
  int b = n >> 11;
  float h[16], f[16];
  for (int e = 0; e < 16; ++e) h[e] = x[(size_t)n * 16 + e] * s[b * 16 + e];
  for (int o = 0; o < 16; ++o) {
    float acc = 0.f;
    for (int e = 0; e < 16; ++e) acc += h[e] * gat_W[o * 16 + e];
    f[o] = acc;
    feat[(size_t)n * 16 + o] = acc;
    accum[(size_t)n * 16 + o] = 0.f;
  }
  for (int hd = 0; hd < 4; ++hd) {
    float al = 0.f, ar = 0.f;
    for (int d = 0; d < 4; ++d) {
      al += f[hd * 4 + d] * attn_l[hd * 4 + d];
      ar += f[hd * 4 + d] * attn_r[hd * 4 + d];
    }
    el[n * 4 + hd] = al;
    er[n * 4 + hd] = ar;
    mbuf[n * 4 + hd] = FMAP_NEG_INF;
    denom[n * 4 + hd] = 0.f;
  }
}

// ---------------- Kernel 4: segment max over dst (per head)
__global__ __launch_bounds__(256) void k_edge_max(const int* __restrict__ src,
    const int* __restrict__ dst, const float* __restrict__ el,
    const float* __restrict__ er, unsigned* __restrict__ mbuf) {
  int e = blockIdx.x * blockDim.x + threadIdx.x;
  if (e >= NE_) return;
  int s = src[e], d = dst[e];
  for (int hd = 0; hd < 4; ++hd) {
    float v = el[s * 4 + hd] + er[d * 4 + hd];
    v = v > 0.f ? v : 0.2f * v;
    atomicMax(&mbuf[d * 4 + hd], fmap(v));
  }
}

// ---------------- Kernel 5: exp / weighted aggregate (unnormalized)
__global__ __launch_bounds__(256) void k_edge_sum(const int* __restrict__ src,
    const int* __restrict__ dst, const float* __restrict__ el,
    const float* __restrict__ er, const unsigned* __restrict__ mbuf,
    const float* __restrict__ feat, float* __restrict__ denom,
    float* __restrict__ accum) {
  int e = blockIdx.x * blockDim.x + threadIdx.x;
  if (e >= NE_) return;
  int s = src[e], d = dst[e];
  for (int hd = 0; hd < 4; ++hd) {
    float v = el[s * 4 + hd] + er[d * 4 + hd];
    v = v > 0.f ? v : 0.2f * v;
    float m = funmap(mbuf[d * 4 + hd]);
    float ee = __expf(v - m);
    atomicAdd(&denom[d * 4 + hd], ee);
    for (int dd = 0; dd < 4; ++dd)
      atomicAdd(&accum[(size_t)d * 16 + hd * 4 + dd],
                ee * feat[(size_t)s * 16 + hd * 4 + dd]);
  }
}

// ---------------- Kernel 6: normalize, +gat_b, project to ha/hb
__global__ __launch_bounds__(256) void k_encode(const float* __restrict__ accum,
    const float* __restrict__ denom, const float* __restrict__ gat_b,
    const float* __restrict__ proj_W, const float* __restrict__ proj_b,
    const float* __restrict__ graph_W, float* __restrict__ encode,
    float* __restrict__ ha, float* __restrict__ hb) {
  int n = blockIdx.x * blockDim.x + threadIdx.x;
  if (n >= N_) return;
  float enc[16];
  for (int hd = 0; hd < 4; ++hd) {
    float den = denom[n * 4 + hd];
    float inv = den > 0.f ? 1.0f / den : 0.f;
    for (int dd = 0; dd < 4; ++dd) {
      int o = hd * 4 + dd;
      float v = accum[(size_t)n * 16 + o] * inv + gat_b[o];
      enc[o] = v;
      encode[(size_t)n * 16 + o] = v;
    }
  }
  float a = 0.f, bb = 0.f;
  for (int g = 0; g < 4; ++g) {
    float hp = proj_b[g];
    for (int o = 0; o < 16; ++o) hp += enc[o] * proj_W[g * 16 + o];
    a  += hp * graph_W[g];
    bb += hp * graph_W[4 + g];
  }
  ha[n] = a;
  hb[n] = bb;
}

// ---------------- Kernel 7: out = w*(encode@encode^T) + (1-w)*(ha[i]+hb[j])
// 64x64 block tile, 16 waves, each wave: one 16x16 WMMA tile (K=16, 4 WMMAs)
__global__ __launch_bounds__(512) void k_final(const float* __restrict__ encode,
    const float* __restrict__ ha, const float* __restrict__ hb,
    const float* __restrict__ w, float* __restrict__ out) {
  __shared__ float As[64][17];
  __shared__ float Bs[64][17];
  int b  = blockIdx.z;
  int i0 = blockIdx.y * 64;
  int j0 = blockIdx.x * 64;
  int t  = threadIdx.x;
  int r  = t >> 4, cc = t & 15;
  size_t basei = ((size_t)b * C_ + i0) * EO;
  size_t basej = ((size_t)b * C_ + j0) * EO;
  As[r][cc]      = encode[basei + (size_t)r * EO + cc];
  As[r + 32][cc] = encode[basei + (size_t)(r + 32) * EO + cc];
  Bs[r][cc]      = encode[basej + (size_t)r * EO + cc];
  Bs[r + 32][cc] = encode[basej + (size_t)(r + 32) * EO + cc];
  __syncthreads();
  int wave = t >> 5;
  int lane = t & 31;
  int ti = wave >> 2, tj = wave & 3;
  int M  = lane & 15, hi = lane >> 4;
  v8f c = {};
  for (int kb = 0; kb < EO; kb += 4) {
    int k = kb + 2 * hi;
    v2f a, bf;
    a.x  = As[ti * 16 + M][k];  a.y  = As[ti * 16 + M][k + 1];
    bf.x = Bs[tj * 16 + M][k];  bf.y = Bs[tj * 16 + M][k + 1]; // N == lane&15
    c = __builtin_amdgcn_wmma_f32_16x16x4_f32(false, a, false, bf,
                                              (short)0, c, false, false);
  }
  int j  = j0 + tj * 16 + M;
  int gj = b * C_ + j;
  float hbv = hb[gj];
  for (int v = 0; v < 8; ++v) {
    int i  = i0 + ti * 16 + v + 8 * hi;
    int gi = b * C_ + i;
    float wv = w[(size_t)i * C_ + j];
    float g2 = ha[gi] + hbv;
    out[((size_t)b * C_ + i) * C_ + j] = wv * c[v] + (1.0f - wv) * g2;
  }
}

extern "C" void kernel_launch(void* const* d_in, const int* in_sizes, int n_in,
                              void* d_out, int out_size, void* d_ws, size_t ws_size,
                              hipStream_t stream) {
  (void)in_sizes; (void)n_in; (void)out_size; (void)ws_size;
  const float* feature = (const float*)d_in[0];
  const int*   src     = (const int*)  d_in[1];
  const int*   dst     = (const int*)  d_in[2];
  const float* emb_W   = (const float*)d_in[3];
  const float* emb_b   = (const float*)d_in[4];
  const float* se_W1   = (const float*)d_in[5];
  const float* se_b1   = (const float*)d_in[6];
  const float* se_W2   = (const float*)d_in[7];
  const float* se_b2   = (const float*)d_in[8];
  const float* gat_W   = (const float*)d_in[9];
  const float* attn_l  = (const float*)d_in[10];
  const float* attn_r  = (const float*)d_in[11];
  const float* gat_b   = (const float*)d_in[12];
  const float* proj_W  = (const float*)d_in[13];
  const float* proj_b  = (const float*)d_in[14];
  const float* graph_W = (const float*)d_in[15];
  const float* w       = (const float*)d_in[16];

  float* ws = (float*)d_ws;
  float*    x      = ws;                 // N*16        = 524288
  float*    zsum   = ws + 524288;        // 256
  float*    s      = ws + 524544;        // 256
  float*    feat   = ws + 524800;        // N*16
  float*    el     = ws + 1049088;       // N*4
  float*    er     = ws + 1180160;       // N*4
  unsigned* mbuf   = (unsigned*)(ws + 1311232); // N*4
  float*    denom  = ws + 1442304;       // N*4
  float*    accum  = ws + 1573376;       // N*16
  float*    encode = ws + 2097664;       // N*16
  float*    ha     = ws + 2621952;       // N
  float*    hb     = ws + 2654720;       // N
  float*    outp   = (float*)d_out;

  hipMemsetAsync(zsum, 0, B_ * EO * sizeof(float), stream);
  k_embed   <<<N_ / 16 / 8, 256, 0, stream>>>(feature, emb_W, emb_b, x, zsum);
  k_se      <<<1, 256, 0, stream>>>(zsum, se_W1, se_b1, se_W2, se_b2, s);
  k_feat    <<<N_ / 256, 256, 0, stream>>>(x, s, gat_W, attn_l, attn_r,
                                           feat, el, er, mbuf, denom, accum);
  k_edge_max<<<NE_ / 256, 256, 0, stream>>>(src, dst, el, er, mbuf);
  k_edge_sum<<<NE_ / 256, 256, 0, stream>>>(src, dst, el, er, mbuf, feat,
                                            denom, accum);
  k_encode  <<<N_ / 256, 256, 0, stream>>>(accum, denom, gat_b, proj_W, proj_b,
                                           graph_W, encode, ha, hb);
  dim3 g7(C_ / 64, C_ / 64, B_);
  k_final   <<<g7, 512, 0, stream>>>(encode, ha, hb, w, outp);
}